// KGAT_49993419325916
// MI455X (gfx1250) — compile-verified
//
#include <hip/hip_runtime.h>
#include <math.h>

#define N_USERS 50000
#define N_ENT   150000
#define N_NODES (N_USERS + N_ENT)
#define E_EDGES 2000000
#define BATCH   8192

typedef float v2f __attribute__((ext_vector_type(2)));
typedef float v8f __attribute__((ext_vector_type(8)));

__device__ __forceinline__ float leaky(float x) { return x > 0.f ? x : 0.01f * x; }

// ---------------------------------------------------------------- utilities
__global__ void zero_f32(float* __restrict__ p, long n) {
  long i = (long)blockIdx.x * blockDim.x + threadIdx.x;
  if (i < n) p[i] = 0.f;
}

// ego = concat(user_embed, entity_embed) along axis 0, 64 dims, float4 vectorized
__global__ void build_ego(const float* __restrict__ ue, const float* __restrict__ ee,
                          float* __restrict__ ego) {
  long i = (long)blockIdx.x * blockDim.x + threadIdx.x;   // float4 index
  long total = (long)N_NODES * 16;                        // 64 floats = 16 float4 per row
  if (i >= total) return;
  long node = i >> 4;
  int  c4   = (int)(i & 15);
  float4 v = (node < N_USERS)
           ? ((const float4*)ue)[node * 16 + c4]
           : ((const float4*)ee)[(node - N_USERS) * 16 + c4];
  ((float4*)ego)[i] = v;
}

// ---------------------------------------------------------------- weight pre-pack
// Repack W[DIN][DOUT] into per-lane B-fragment order so each lane's fragment is
// one contiguous float2: Wp[tn][kq][lane] = {W[ka][n], W[ka+1][n]},
// ka = kq*4 + (lane>>4)*2, n = tn*16 + (lane&15).
template<int DIN, int DOUT>
__global__ void pack_w(const float* __restrict__ W, float* __restrict__ Wp) {
  int i = blockIdx.x * blockDim.x + threadIdx.x;
  const int total = (DOUT / 16) * (DIN / 4) * 32;
  if (i >= total) return;
  int lane = i & 31;
  int rest = i >> 5;
  int kq = rest % (DIN / 4);
  int tn = rest / (DIN / 4);
  int half = lane >> 4, lr = lane & 15;
  int ka = kq * 4 + half * 2;
  int n  = tn * 16 + lr;
  v2f w = {W[ka * DOUT + n], W[(ka + 1) * DOUT + n]};
  ((v2f*)Wp)[i] = w;
}

// ---------------------------------------------------------------- SpMM scatter
// side[rows[e]] += vals[e] * ego[cols[e]]   (per 16-dim chunk per thread)
template<int DIN>
__global__ void spmm_scatter(const int* __restrict__ rows, const int* __restrict__ cols,
                             const float* __restrict__ vals,
                             const float* __restrict__ ego, float* __restrict__ side) {
  const int CH = DIN / 16;
  long t = (long)blockIdx.x * blockDim.x + threadIdx.x;
  long total = (long)E_EDGES * CH;
  if (t >= total) return;
  long e = t / CH;
  int chunk = (int)(t - e * CH);
  int r = rows[e], c = cols[e];
  float v = vals[e];
  const float* src = ego  + (long)c * DIN + chunk * 16;
  float*       dst = side + (long)r * DIN + chunk * 16;
#pragma unroll
  for (int j = 0; j < 16; j += 4) {
    float4 x = *(const float4*)(src + j);
    __hip_atomic_fetch_add(dst + j + 0, v * x.x, __ATOMIC_RELAXED, __HIP_MEMORY_SCOPE_AGENT);
    __hip_atomic_fetch_add(dst + j + 1, v * x.y, __ATOMIC_RELAXED, __HIP_MEMORY_SCOPE_AGENT);
    __hip_atomic_fetch_add(dst + j + 2, v * x.z, __ATOMIC_RELAXED, __HIP_MEMORY_SCOPE_AGENT);
    __hip_atomic_fetch_add(dst + j + 3, v * x.w, __ATOMIC_RELAXED, __HIP_MEMORY_SCOPE_AGENT);
  }
}

// ---------------------------------------------------------------- fused dual GEMM (WMMA f32 16x16x4)
// out = leaky((ego+side)@Wg + bg) + leaky((ego*side)@Wb + bb)
// One wave owns a full 16-row slab and register-blocks ALL DOUT/16 n-tiles:
// A fragments loaded once per k-step, reused across tiles; weights come from
// the packed fragment buffer as single b64 loads.
template<int DIN, int DOUT>
__global__ void __launch_bounds__(256)
layer_gemm(const float* __restrict__ ego, const float* __restrict__ side,
           const float* __restrict__ Wgp, const float* __restrict__ bg,
           const float* __restrict__ Wbp, const float* __restrict__ bb,
           float* __restrict__ out) {
  const int NT = DOUT / 16;               // n-tiles, register-blocked
  const int KQ = DIN / 4;                 // k-steps
  int wid  = (int)(((long)blockIdx.x * blockDim.x + threadIdx.x) >> 5);
  int lane = threadIdx.x & 31;
  if (wid >= N_NODES / 16) return;        // wave-uniform: EXEC stays all-ones
  int half = lane >> 4, lr = lane & 15;

  const float* egoRow  = ego  + (long)(wid * 16 + lr) * DIN;
  const float* sideRow = side + (long)(wid * 16 + lr) * DIN;
  const v2f* WgF = (const v2f*)Wgp;       // [NT][KQ][32]
  const v2f* WbF = (const v2f*)Wbp;

  v8f cg[NT], cb[NT];
#pragma unroll
  for (int t = 0; t < NT; ++t) {
    cg[t] = (v8f){0.f,0.f,0.f,0.f,0.f,0.f,0.f,0.f};
    cb[t] = (v8f){0.f,0.f,0.f,0.f,0.f,0.f,0.f,0.f};
  }

#pragma unroll
  for (int kq = 0; kq < KQ; ++kq) {
    int ka = kq * 4 + half * 2;           // this lane's two K values
    float e0 = egoRow[ka],  e1 = egoRow[ka + 1];
    float s0 = sideRow[ka], s1 = sideRow[ka + 1];
    v2f aH = {e0 + s0, e1 + s1};          // (ego + side) fragment
    v2f aB = {e0 * s0, e1 * s1};          // (ego * side) fragment
#pragma unroll
    for (int t = 0; t < NT; ++t) {
      v2f bG = WgF[(t * KQ + kq) * 32 + lane];
      v2f bB = WbF[(t * KQ + kq) * 32 + lane];
      cg[t] = __builtin_amdgcn_wmma_f32_16x16x4_f32(false, aH, false, bG, (short)0, cg[t], false, false);
      cb[t] = __builtin_amdgcn_wmma_f32_16x16x4_f32(false, aB, false, bB, (short)0, cb[t], false, false);
    }
  }

#pragma unroll
  for (int t = 0; t < NT; ++t) {
    int n = t * 16 + lr;
    float biasg = bg[n], biasb = bb[n];
#pragma unroll
    for (int r = 0; r < 8; ++r) {
      int m = wid * 16 + r + half * 8;    // C/D layout: lanes>=16 hold rows +8
      float vg = leaky(cg[t][r] + biasg);
      float vb = leaky(cb[t][r] + biasb);
      out[(long)m * DOUT + n] = vg + vb;
    }
  }
}

// ---------------------------------------------------------------- row L2 norm (wave per row)
template<int D>
__global__ void l2norm_rows(const float* __restrict__ x, float* __restrict__ y) {
  int wid  = (int)(((long)blockIdx.x * blockDim.x + threadIdx.x) >> 5);
  int lane = threadIdx.x & 31;
  if (wid >= N_NODES) return;
  const float* row = x + (long)wid * D;
  float ss = 0.f;
  for (int d = lane; d < D; d += 32) { float v = row[d]; ss += v * v; }
#pragma unroll
  for (int o = 16; o > 0; o >>= 1) ss += __shfl_xor(ss, o, 32);
  float inv = 1.f / fmaxf(sqrtf(ss), 1e-12f);
  float* orow = y + (long)wid * D;
  for (int d = lane; d < D; d += 32) orow[d] = row[d] * inv;
}

// ---------------------------------------------------------------- batch scoring (wave per pair)
__global__ void score_kernel(const int* __restrict__ users, const int* __restrict__ pos,
                             const int* __restrict__ neg,
                             const float* __restrict__ ue, const float* __restrict__ ee,
                             const float* __restrict__ n1, const float* __restrict__ n2,
                             const float* __restrict__ n3, float* __restrict__ out) {
  int wid  = (int)(((long)blockIdx.x * blockDim.x + threadIdx.x) >> 5);
  int lane = threadIdx.x & 31;
  if (wid >= BATCH) return;
  long u = users[wid];
  long p = pos[wid], g = neg[wid];
  long pu = (long)N_USERS + p, gu = (long)N_USERS + g;
  float dp = 0.f, dn = 0.f;
  // segment 0: original (un-normalized) embeddings, 64 dims
  for (int d = lane; d < 64; d += 32) {
    float uv = ue[u * 64 + d];
    dp += uv * ee[p * 64 + d];
    dn += uv * ee[g * 64 + d];
  }
  // segment 1: layer-1 normalized, 64 dims
  for (int d = lane; d < 64; d += 32) {
    float uv = n1[u * 64 + d];
    dp += uv * n1[pu * 64 + d];
    dn += uv * n1[gu * 64 + d];
  }
  // segment 2: layer-2 normalized, 32 dims
  {
    float uv = n2[u * 32 + lane];
    dp += uv * n2[pu * 32 + lane];
    dn += uv * n2[gu * 32 + lane];
  }
  // segment 3: layer-3 normalized, 16 dims
  if (lane < 16) {
    float uv = n3[u * 16 + lane];
    dp += uv * n3[pu * 16 + lane];
    dn += uv * n3[gu * 16 + lane];
  }
#pragma unroll
  for (int o = 16; o > 0; o >>= 1) { dp += __shfl_xor(dp, o, 32); dn += __shfl_xor(dn, o, 32); }
  if (lane == 0) { out[wid * 2 + 0] = dp; out[wid * 2 + 1] = dn; }
}

// ---------------------------------------------------------------- launch
static inline int nblk(long n, int t) { return (int)((n + t - 1) / t); }

extern "C" void kernel_launch(void* const* d_in, const int* in_sizes, int n_in,
                              void* d_out, int out_size, void* d_ws, size_t ws_size,
                              hipStream_t stream) {
  (void)in_sizes; (void)n_in; (void)out_size; (void)ws_size;
  const int*   users = (const int*)d_in[0];
  const int*   pos   = (const int*)d_in[1];
  const int*   neg   = (const int*)d_in[2];
  const int*   rows  = (const int*)d_in[3];
  const int*   cols  = (const int*)d_in[4];
  const float* evals = (const float*)d_in[5];
  const float* ue    = (const float*)d_in[6];
  const float* ee    = (const float*)d_in[7];
  const float* Wg0 = (const float*)d_in[8],  *bg0 = (const float*)d_in[9];
  const float* Wb0 = (const float*)d_in[10], *bb0 = (const float*)d_in[11];
  const float* Wg1 = (const float*)d_in[12], *bg1 = (const float*)d_in[13];
  const float* Wb1 = (const float*)d_in[14], *bb1 = (const float*)d_in[15];
  const float* Wg2 = (const float*)d_in[16], *bg2 = (const float*)d_in[17];
  const float* Wb2 = (const float*)d_in[18], *bb2 = (const float*)d_in[19];
  float* out = (float*)d_out;

  const long NB = N_NODES;
  float* ws   = (float*)d_ws;
  float* egoA = ws;                 // N x 64
  float* egoB = egoA + NB * 64;     // N x 64
  float* side = egoB + NB * 64;     // N x 64 (reused per layer)
  float* n1   = side + NB * 64;     // N x 64
  float* n2   = n1   + NB * 64;     // N x 32
  float* n3   = n2   + NB * 32;     // N x 16
  float* wgp0 = n3   + NB * 16;     // packed weights
  float* wbp0 = wgp0 + 64 * 64;
  float* wgp1 = wbp0 + 64 * 64;
  float* wbp1 = wgp1 + 64 * 32;
  float* wgp2 = wbp1 + 64 * 32;
  float* wbp2 = wgp2 + 32 * 16;

  const int T = 256;

  build_ego<<<nblk(NB * 16, T), T, 0, stream>>>(ue, ee, egoA);

  // one-time weight fragment repack (tiny)
  pack_w<64, 64><<<nblk(64 * 64 / 2, T), T, 0, stream>>>(Wg0, wgp0);
  pack_w<64, 64><<<nblk(64 * 64 / 2, T), T, 0, stream>>>(Wb0, wbp0);
  pack_w<64, 32><<<nblk(64 * 32 / 2, T), T, 0, stream>>>(Wg1, wgp1);
  pack_w<64, 32><<<nblk(64 * 32 / 2, T), T, 0, stream>>>(Wb1, wbp1);
  pack_w<32, 16><<<nblk(32 * 16 / 2, T), T, 0, stream>>>(Wg2, wgp2);
  pack_w<32, 16><<<nblk(32 * 16 / 2, T), T, 0, stream>>>(Wb2, wbp2);

  // ---- layer 0: 64 -> 64, in egoA, out egoB
  zero_f32<<<nblk(NB * 64, T), T, 0, stream>>>(side, NB * 64);
  spmm_scatter<64><<<nblk((long)E_EDGES * 4, T), T, 0, stream>>>(rows, cols, evals, egoA, side);
  layer_gemm<64, 64><<<nblk((long)(N_NODES / 16) * 32, T), T, 0, stream>>>(
      egoA, side, wgp0, bg0, wbp0, bb0, egoB);
  l2norm_rows<64><<<nblk(NB * 32, T), T, 0, stream>>>(egoB, n1);

  // ---- layer 1: 64 -> 32, in egoB, out egoA
  zero_f32<<<nblk(NB * 64, T), T, 0, stream>>>(side, NB * 64);
  spmm_scatter<64><<<nblk((long)E_EDGES * 4, T), T, 0, stream>>>(rows, cols, evals, egoB, side);
  layer_gemm<64, 32><<<nblk((long)(N_NODES / 16) * 32, T), T, 0, stream>>>(
      egoB, side, wgp1, bg1, wbp1, bb1, egoA);
  l2norm_rows<32><<<nblk(NB * 32, T), T, 0, stream>>>(egoA, n2);

  // ---- layer 2: 32 -> 16, in egoA, out egoB
  zero_f32<<<nblk(NB * 32, T), T, 0, stream>>>(side, NB * 32);
  spmm_scatter<32><<<nblk((long)E_EDGES * 2, T), T, 0, stream>>>(rows, cols, evals, egoA, side);
  layer_gemm<32, 16><<<nblk((long)(N_NODES / 16) * 32, T), T, 0, stream>>>(
      egoA, side, wgp2, bg2, wbp2, bb2, egoB);
  l2norm_rows<16><<<nblk(NB * 32, T), T, 0, stream>>>(egoB, n3);

  // ---- scoring
  score_kernel<<<nblk((long)BATCH * 32, T), T, 0, stream>>>(users, pos, neg, ue, ee, n1, n2, n3, out);
}